// FlexiblePatchSelector_75806172774840
// MI455X (gfx1250) — compile-verified
//
#include <hip/hip_runtime.h>
#include <stdint.h>

// Problem constants (fixed by the reference): B=64, N=4096, D=768, k=N/4=1024.
#define BB   64
#define NN   4096
#define DD   768
#define KK   1024

#define ROWS_PER_BLOCK 8
#define THREADS2 192          // DD floats / 4 per float4 = 192 lanes (6 waves of 32)

typedef float v4f __attribute__((ext_vector_type(4)));

// ---------------------------------------------------------------------------
// Helpers
// ---------------------------------------------------------------------------

// Map float bits to a monotonically increasing unsigned key (ascending float
// order). We invert it so ascending-uint sort == descending-float sort.
__device__ __forceinline__ uint32_t mono_f32(float f) {
  uint32_t u = __float_as_uint(f);
  return (u & 0x80000000u) ? ~u : (u | 0x80000000u);
}

// Low 32 bits of a flat pointer into the LDS aperture == LDS byte address
// (flat = aperture_hi32 << 32 | lds_offset32 on AMDGPU).
__device__ __forceinline__ uint32_t lds_off(const void* p) {
  return (uint32_t)(uintptr_t)p;
}

// Async DMA: copy 16 bytes per active lane from global memory into LDS.
// Tracked by ASYNCcnt; dst LDS byte address per lane in a VGPR, global address
// in a VGPR pair, GV addressing mode (saddr = off). RT (default) temporal hint:
// used for the pos-embed table, which is re-read 64x and should stay in L2.
__device__ __forceinline__ void async_copy_b128(uint32_t lds_byte, const void* gptr) {
  asm volatile("global_load_async_to_lds_b128 %0, %1, off"
               :: "v"(lds_byte), "v"(gptr)
               : "memory");
}

// Same, but non-temporal: used for patch rows, which are read exactly once —
// keep them from evicting the L2-resident pos-embed table.
__device__ __forceinline__ void async_copy_b128_nt(uint32_t lds_byte, const void* gptr) {
  asm volatile("global_load_async_to_lds_b128 %0, %1, off th:TH_LOAD_NT"
               :: "v"(lds_byte), "v"(gptr)
               : "memory");
}

// ---------------------------------------------------------------------------
// Kernel: bandwidth kernel. Gather selected patch rows + pos-embed rows via
// async global->LDS DMA, double-buffered and software-pipelined; add; store NT.
// Each lane consumes exactly the 16 LDS bytes it DMA'd, so the only sync
// needed is the wave-private ASYNCcnt / DScnt counters (no barriers).
// ---------------------------------------------------------------------------
__global__ __launch_bounds__(THREADS2) void gather_add_kernel(
    const float* __restrict__ patches,   // (B, N, D)
    const float* __restrict__ pe,        // (1, N+1, D); row 0 skipped
    const int*   __restrict__ sel,       // (B, K) indices from top-k kernel
    float*       __restrict__ out) {     // (B, K, D)
  __shared__ v4f bufP[2][THREADS2];
  __shared__ v4f bufE[2][THREADS2];

  const int  t    = threadIdx.x;
  const long row0 = (long)blockIdx.x * ROWS_PER_BLOCK;

  const uint32_t lP[2] = { lds_off(&bufP[0][t]), lds_off(&bufP[1][t]) };
  const uint32_t lE[2] = { lds_off(&bufE[0][t]), lds_off(&bufE[1][t]) };

  // Uniform (scalar) preload of this block's 8 gather indices.
  int pidx[ROWS_PER_BLOCK];
#pragma unroll
  for (int r = 0; r < ROWS_PER_BLOCK; ++r) pidx[r] = sel[row0 + r];

  auto issue = [&](int r) {
    long row = row0 + r;
    int  bb  = (int)(row >> 10);                    // row / K
    const float* gp = patches + ((size_t)bb * NN + (size_t)pidx[r]) * DD + t * 4;
    const float* ge = pe + ((size_t)(pidx[r] + 1)) * DD + t * 4;  // skip CLS row
    async_copy_b128_nt(lP[r & 1], gp);   // read-once: NT
    async_copy_b128(lE[r & 1], ge);      // reused across batches: RT
  };

  issue(0);
  issue(1);

#pragma unroll
  for (int r = 0; r < ROWS_PER_BLOCK; ++r) {
    // Async loads complete in order: <=2 outstanding means row r's two copies
    // have landed in LDS (row r+1's pair may still be in flight).
    if (r + 1 < ROWS_PER_BLOCK)
      asm volatile("s_wait_asynccnt 0x2" ::: "memory");
    else
      asm volatile("s_wait_asynccnt 0x0" ::: "memory");

    v4f p = bufP[r & 1][t];
    v4f e = bufE[r & 1][t];
    v4f o = p + e;

    long row = row0 + r;
    // Output is write-once, never re-read: non-temporal store keeps it from
    // flushing the pos-embed table out of L2.
    __builtin_nontemporal_store(o, (v4f*)(out + (size_t)row * DD + (size_t)t * 4));

    if (r + 2 < ROWS_PER_BLOCK) {
      // WAR: make sure our ds_loads from this buffer have returned before the
      // DMA engine overwrites it with row r+2.
      asm volatile("s_wait_dscnt 0x0" ::: "memory");
      issue(r + 2);
    }
  }
}

// ---------------------------------------------------------------------------
// Kernel: exact top-k per batch row via in-LDS bitonic sort of 64-bit keys.
// key = (~mono(score) << 32) | index  -> ascending sort gives descending score,
// ties broken by ascending index (matches jax.lax.top_k).
// One 1024-thread workgroup per batch row; 32 KB LDS for 4096 keys.
// ---------------------------------------------------------------------------
__global__ __launch_bounds__(1024) void topk_sort_kernel(
    const float* __restrict__ scores, int* __restrict__ idx_out) {
  __shared__ uint64_t keys[NN];
  const int b = blockIdx.x;
  const int t = threadIdx.x;
  const float* s = scores + (size_t)b * NN;

  for (int i = t; i < NN; i += 1024) {
    uint32_t m = ~mono_f32(s[i]);                 // descending-score key
    keys[i] = ((uint64_t)m << 32) | (uint32_t)i;  // stable tie-break on index
  }

  for (int kk = 2; kk <= NN; kk <<= 1) {
    for (int j = kk >> 1; j > 0; j >>= 1) {
      __syncthreads();
      for (int p = t; p < NN / 2; p += 1024) {
        int i   = ((p & ~(j - 1)) << 1) | (p & (j - 1));
        int ixj = i | j;
        uint64_t a = keys[i];
        uint64_t c = keys[ixj];
        bool up = ((i & kk) == 0);
        if (up ? (a > c) : (a < c)) { keys[i] = c; keys[ixj] = a; }
      }
    }
  }
  __syncthreads();

  if (t < KK) idx_out[(size_t)b * KK + t] = (int)(keys[t] & 0xFFFFFFFFu);
}

// ---------------------------------------------------------------------------
// Launch
// ---------------------------------------------------------------------------
extern "C" void kernel_launch(void* const* d_in, const int* in_sizes, int n_in,
                              void* d_out, int out_size, void* d_ws, size_t ws_size,
                              hipStream_t stream) {
  const float* patches = (const float*)d_in[0];   // (64, 4096, 768) f32
  const float* pe      = (const float*)d_in[1];   // (1, 4097, 768)  f32
  const float* scores  = (const float*)d_in[2];   // (64, 4096)      f32
  float* out = (float*)d_out;                     // (64, 1024, 768) f32
  int*   sel = (int*)d_ws;                        // 64*1024 ints scratch

  topk_sort_kernel<<<BB, 1024, 0, stream>>>(scores, sel);

  const int total_rows = BB * KK;                 // 65536
  gather_add_kernel<<<total_rows / ROWS_PER_BLOCK, THREADS2, 0, stream>>>(
      patches, pe, sel, out);
}